// ConditionalDecoder_35622458753385
// MI455X (gfx1250) — compile-verified
//
#include <hip/hip_runtime.h>
#include <hip/hip_bf16.h>

typedef __attribute__((ext_vector_type(2))) float v2f;
typedef __attribute__((ext_vector_type(8))) float v8f;

#define B_  16
#define L_  256
#define H_  256
#define G4_ 1024   // 4*H
#define V_  32000
#define M_  4096   // B*L

static __device__ __forceinline__ v8f wmma4(v2f a, v2f b, v8f c) {
  // D = A(16x4,f32) * B(4x16,f32) + C(16x16,f32)
  return __builtin_amdgcn_wmma_f32_16x16x4_f32(
      /*neg_a=*/false, a, /*neg_b=*/false, b,
      /*c_mod=*/(short)0, c, /*reuse_a=*/false, /*reuse_b=*/false);
}

static __device__ __forceinline__ float sigf(float x) {
  return 1.0f / (1.0f + __expf(-x));
}

// Issue one per-lane 16B async copy global -> LDS (ASYNCcnt tracked).
static __device__ __forceinline__ void async_b128(unsigned lds_byte_off,
                                                  const float* gaddr) {
  asm volatile("global_load_async_to_lds_b128 %0, %1, off"
               :: "v"(lds_byte_off), "v"(gaddr) : "memory");
}
static __device__ __forceinline__ void wait_async0() {
  asm volatile("s_wait_asynccnt 0x0" ::: "memory");
}

// ---------------------------------------------------------------------------
// X[m][k] = (tok==0) ? 0 : emb[tok][k];  m = b*L + t, tok = seq[b*(L+1)+t]
// ---------------------------------------------------------------------------
__global__ void embed_gather_kernel(const int* __restrict__ seq,
                                    const float* __restrict__ emb,
                                    float* __restrict__ X) {
  int m = blockIdx.x;
  int k = threadIdx.x;
  int b = m >> 8, t = m & 255;
  int tok = seq[b * (L_ + 1) + t];
  float v = (tok == 0) ? 0.0f : emb[(size_t)tok * H_ + k];
  X[(size_t)m * H_ + k] = v;
}

// ---------------------------------------------------------------------------
// gamma[b][j] = z[b]·Wg[j] + bg[j];  beta likewise
// ---------------------------------------------------------------------------
__global__ void film_params_kernel(const float* __restrict__ z,
                                   const float* __restrict__ Wg,
                                   const float* __restrict__ bg,
                                   const float* __restrict__ Wb,
                                   const float* __restrict__ bb,
                                   float* __restrict__ gamma,
                                   float* __restrict__ beta) {
  int b = blockIdx.x;
  int j = threadIdx.x;
  float sg = 0.f, sb = 0.f;
  for (int k = 0; k < H_; ++k) {
    float zv = z[b * H_ + k];
    sg += zv * Wg[(size_t)j * H_ + k];
    sb += zv * Wb[(size_t)j * H_ + k];
  }
  gamma[b * H_ + j] = sg + bg[j];
  beta [b * H_ + j] = sb + bb[j];
}

// ---------------------------------------------------------------------------
// Generic C[m][n] = sum_k A[m][k]*Bm[n][k] + bias1[n] (+bias2[n])
// Block: 256 thr (8 waves), 64(M)x128(N) tile; wave tile 16x64 (4 acc).
// A/B tiles staged row-major via GLOBAL_LOAD_ASYNC_TO_LDS_B128, stride 68
// floats (bank-conflict-free frag reads, 16B-aligned async chunks).
// M % 64 == 0, N % 128 == 0, K % 64 == 0.
// ---------------------------------------------------------------------------
__global__ void gemm_nt_bias_kernel(const float* __restrict__ A,
                                    const float* __restrict__ Bm,
                                    const float* __restrict__ bias1,
                                    const float* __restrict__ bias2,
                                    float* __restrict__ C,
                                    int M, int N, int K) {
  __shared__ __align__(16) float As[64 * 68];   // [m][k] rows, stride 68
  __shared__ __align__(16) float Bs[128 * 68];  // [n][k] rows, stride 68
  const int m0 = blockIdx.y * 64;
  const int n0 = blockIdx.x * 128;
  const int tid  = threadIdx.x;
  const int lane = tid & 31;
  const int w    = tid >> 5;
  const int wm   = (w >> 1) * 16;  // 0,16,32,48
  const int wn   = (w & 1) * 64;   // 0 or 64
  const int half = lane >> 4;      // 0/1
  const int l16  = lane & 15;

  const unsigned asBase = (unsigned)(uintptr_t)(&As[0]);
  const unsigned bsBase = (unsigned)(uintptr_t)(&Bs[0]);

  const v8f zero8 = {0.f,0.f,0.f,0.f,0.f,0.f,0.f,0.f};
  v8f acc[4];
#pragma unroll
  for (int s = 0; s < 4; ++s) acc[s] = zero8;

  for (int k0 = 0; k0 < K; k0 += 64) {
    // stage: A tile 64x64 (256 x 16B) + B tile 128x64 (512 x 16B) = 768 chunks
#pragma unroll
    for (int ii = 0; ii < 3; ++ii) {
      int idx = tid + ii * 256;
      if (idx < 256) {
        int m = idx >> 2, kq = (idx & 3) << 4;
        async_b128(asBase + (unsigned)(m * 68 + kq) * 4u,
                   A + (size_t)(m0 + m) * K + k0 + kq);
      } else {
        int j = idx - 256;
        int n = j >> 2, kq = (j & 3) << 4;
        async_b128(bsBase + (unsigned)(n * 68 + kq) * 4u,
                   Bm + (size_t)(n0 + n) * K + k0 + kq);
      }
    }
    wait_async0();
    __syncthreads();

#pragma unroll 4
    for (int kk = 0; kk < 64; kk += 4) {
      int ka = kk + half * 2;
      const float* ap = &As[(wm + l16) * 68 + ka];
      v2f a = { ap[0], ap[1] };
#pragma unroll
      for (int s = 0; s < 4; ++s) {
        const float* bp = &Bs[(wn + s * 16 + l16) * 68 + ka];
        v2f b = { bp[0], bp[1] };
        acc[s] = wmma4(a, b, acc[s]);
      }
    }
    __syncthreads();
  }

  // epilogue: VGPR i -> row = i + half*8, col lane-mapped
#pragma unroll
  for (int s = 0; s < 4; ++s) {
    int cn = n0 + wn + s * 16 + l16;
    float ba = (bias1 ? bias1[cn] : 0.f) + (bias2 ? bias2[cn] : 0.f);
#pragma unroll
    for (int i = 0; i < 8; ++i) {
      int row = m0 + wm + i + half * 8;
      C[(size_t)row * N + cn] = acc[s][i] + ba;
    }
  }
}

// ---------------------------------------------------------------------------
// LSTM scan: 1 block, 1024 threads (32 waves). G = x@Wih^T + b_ih + b_hh
// precomputed for all t. Per step: gates += h@Whh^T, then gate nonlinearity.
// Wave pair p owns hidden slice j in [16p,16p+16): even wave -> i,f tiles,
// odd wave -> g,o tiles (exchanged via LDS). Cell state lives in even-wave
// registers. Batch (16) == WMMA M dimension.
// ---------------------------------------------------------------------------
__global__ void lstm_scan_kernel(const float* __restrict__ G,    // [B*L][4H]
                                 const float* __restrict__ Whh,  // [4H][H]
                                 float* __restrict__ Hout) {     // [B*L][H]
  __shared__ float h_lds[16 * 260];          // [b][k], stride 260 (conflict-free)
  __shared__ float go_lds[2 * 16 * 16 * 16]; // [g/o][pair][b][jc]
  const int tid  = threadIdx.x;
  const int lane = tid & 31;
  const int w    = tid >> 5;
  const int pair = w >> 1;      // 0..15 == j-tile
  const int odd  = w & 1;
  const int half = lane >> 4;
  const int l16  = lane & 15;

  for (int i = tid; i < 16 * 260; i += 1024) h_lds[i] = 0.f;
  float cst[8];
#pragma unroll
  for (int i = 0; i < 8; ++i) cst[i] = 0.f;
  __syncthreads();

  const int nA = pair * 16 + (odd ? 2 * H_ : 0);      // i-gate or g-gate cols
  const int nB = pair * 16 + (odd ? 3 * H_ : H_);     // f-gate or o-gate cols

#pragma unroll 1
  for (int t = 0; t < L_; ++t) {
    v8f accA = {0.f,0.f,0.f,0.f,0.f,0.f,0.f,0.f};
    v8f accB = {0.f,0.f,0.f,0.f,0.f,0.f,0.f,0.f};
#pragma unroll 4
    for (int kk = 0; kk < H_; kk += 4) {
      int ka = kk + half * 2;
      v2f a = { h_lds[l16 * 260 + ka], h_lds[l16 * 260 + ka + 1] };
      const float* pA = Whh + (size_t)(nA + l16) * H_ + ka;
      const float* pB = Whh + (size_t)(nB + l16) * H_ + ka;
      v2f vb0 = { pA[0], pA[1] };
      v2f vb1 = { pB[0], pB[1] };
      accA = wmma4(a, vb0, accA);
      accB = wmma4(a, vb1, accB);
    }
    if (odd) {
#pragma unroll
      for (int i = 0; i < 8; ++i) {
        int b = i + half * 8;
        size_t row = ((size_t)b * L_ + t) * G4_;
        go_lds[(pair * 16 + b) * 16 + l16]        = accA[i] + G[row + nA + l16]; // g pre
        go_lds[4096 + (pair * 16 + b) * 16 + l16] = accB[i] + G[row + nB + l16]; // o pre
      }
    }
    __syncthreads();
    if (!odd) {
#pragma unroll
      for (int i = 0; i < 8; ++i) {
        int b = i + half * 8;
        size_t row = ((size_t)b * L_ + t) * G4_;
        float ig = sigf(accA[i] + G[row + nA + l16]);
        float fg = sigf(accB[i] + G[row + nB + l16]);
        float gg = tanhf(go_lds[(pair * 16 + b) * 16 + l16]);
        float oo = sigf(go_lds[4096 + (pair * 16 + b) * 16 + l16]);
        float cn = fg * cst[i] + ig * gg;
        cst[i] = cn;
        float hn = oo * tanhf(cn);
        int j = pair * 16 + l16;
        h_lds[b * 260 + j] = hn;
        Hout[((size_t)b * L_ + t) * H_ + j] = hn;
      }
    }
    __syncthreads();
  }
}

// ---------------------------------------------------------------------------
// H2[m][j] = gamma[b][j]*H2[m][j] + beta[b][j]   (in place)
// ---------------------------------------------------------------------------
__global__ void film_apply_kernel(float* __restrict__ H2,
                                  const float* __restrict__ gamma,
                                  const float* __restrict__ beta) {
  size_t idx = (size_t)blockIdx.x * 256 + threadIdx.x;
  int j = (int)(idx & 255);
  int b = (int)(idx >> 16);  // m = idx>>8, b = m>>8
  H2[idx] = gamma[b * H_ + j] * H2[idx] + beta[b * H_ + j];
}

// ---------------------------------------------------------------------------
extern "C" void kernel_launch(void* const* d_in, const int* in_sizes, int n_in,
                              void* d_out, int out_size, void* d_ws, size_t ws_size,
                              hipStream_t stream) {
  const float* z     = (const float*)d_in[0];
  const int*   seq   = (const int*)  d_in[1];
  const float* emb   = (const float*)d_in[2];
  const float* Wg    = (const float*)d_in[3];
  const float* bg    = (const float*)d_in[4];
  const float* Wb    = (const float*)d_in[5];
  const float* bb    = (const float*)d_in[6];
  const float* Wih0  = (const float*)d_in[7];
  const float* Whh0  = (const float*)d_in[8];
  const float* bih0  = (const float*)d_in[9];
  const float* bhh0  = (const float*)d_in[10];
  const float* Wih1  = (const float*)d_in[11];
  const float* Whh1  = (const float*)d_in[12];
  const float* bih1  = (const float*)d_in[13];
  const float* bhh1  = (const float*)d_in[14];
  const float* Wout  = (const float*)d_in[15];
  const float* bout  = (const float*)d_in[16];
  float* out = (float*)d_out;

  char* ws = (char*)d_ws;
  float* X     = (float*)(ws);                          // 4096*256   (4 MB)
  float* G     = (float*)(ws + (size_t)(4  << 20));     // 4096*1024  (16 MB)
  float* H1    = (float*)(ws + (size_t)(20 << 20));     // 4096*256   (4 MB)
  float* H2    = (float*)(ws + (size_t)(24 << 20));     // 4096*256   (4 MB)
  float* gamma = (float*)(ws + (size_t)(28 << 20));     // 16*256
  float* beta  = gamma + B_ * H_;

  // 1) embedding gather (+ zero row 0)
  embed_gather_kernel<<<dim3(M_), dim3(H_), 0, stream>>>(seq, emb, X);

  // 2) FiLM parameters
  film_params_kernel<<<dim3(B_), dim3(H_), 0, stream>>>(z, Wg, bg, Wb, bb, gamma, beta);

  // 3) layer-0 input GEMM: G = X @ Wih0^T + (bih0 + bhh0)
  gemm_nt_bias_kernel<<<dim3(G4_ / 128, M_ / 64), dim3(256), 0, stream>>>(
      X, Wih0, bih0, bhh0, G, M_, G4_, H_);

  // 4) layer-0 recurrent scan -> H1
  lstm_scan_kernel<<<dim3(1), dim3(1024), 0, stream>>>(G, Whh0, H1);

  // 5) layer-1 input GEMM: G = H1 @ Wih1^T + (bih1 + bhh1)
  gemm_nt_bias_kernel<<<dim3(G4_ / 128, M_ / 64), dim3(256), 0, stream>>>(
      H1, Wih1, bih1, bhh1, G, M_, G4_, H_);

  // 6) layer-1 recurrent scan -> H2
  lstm_scan_kernel<<<dim3(1), dim3(1024), 0, stream>>>(G, Whh1, H2);

  // 7) FiLM: H2 = gamma*H2 + beta (in place)
  film_apply_kernel<<<dim3(M_), dim3(256), 0, stream>>>(H2, gamma, beta);

  // 8) output projection: out = H2 @ Wout^T + bout   (4096 x 32000)
  gemm_nt_bias_kernel<<<dim3(V_ / 128, M_ / 64), dim3(256), 0, stream>>>(
      H2, Wout, bout, nullptr, out, M_, V_, H_);
}